// RelPartialLearnableMultiHeadAttn_51754355917020
// MI455X (gfx1250) — compile-verified
//
#include <hip/hip_runtime.h>
#include <math.h>

// ---------------------------------------------------------------------------
// Transformer-XL RelPartialLearnableMultiHeadAttn forward for MI455X (gfx1250)
// Full fp32 pipeline built on V_WMMA_F32_16X16X4_F32 (wave32 WMMA).
//
//  K1: QKV = [mems;w] @ W_qkv            (16384x512 @ 512x1536)   WMMA GEMM
//  K2: RK  = r @ W_r                     (2048x512  @ 512x512)    WMMA GEMM
//  K3: fused flash attention per (iblock,b,n): AC + rel-shifted BD via WMMA,
//      online softmax + streaming entropy, P@V via WMMA
//  K4: out = attn_vec @ W_o              (8192x512  @ 512x512)    WMMA GEMM
//  K5: deterministic entropy finalize (fixed-order reduction, no atomics)
//
// v3: fix async-to-LDS builtin pointer types (expects int4 addrspace(1)/(3)
//     pointers, per hipcc diagnostic). Otherwise identical to v2.
// ---------------------------------------------------------------------------

typedef float v2f __attribute__((ext_vector_type(2)));
typedef float v8f __attribute__((ext_vector_type(8)));
typedef int   v4i __attribute__((ext_vector_type(4)));

#define QLEN   1024
#define MLEN   1024
#define KLEN_  2048
#define BSZ    8
#define NH     8
#define DH     64
#define DMODEL 512
#define SCALE  0.125f     /* 1/sqrt(64) */
#define NEG_BIG (-1e30f)

#if __has_builtin(__builtin_amdgcn_global_load_async_to_lds_b128) && \
    __has_builtin(__builtin_amdgcn_s_wait_asynccnt)
#define ASYNC_LDS 1
#else
#define ASYNC_LDS 0
#endif

#if ASYNC_LDS
typedef __attribute__((address_space(1))) v4i* g4ptr;
typedef __attribute__((address_space(3))) v4i* l4ptr;
#endif

// Copy 16 contiguous bytes global -> LDS (per-lane).
static __device__ __forceinline__ void copy16_to_lds(float* lds, const float* g) {
#if ASYNC_LDS
    __builtin_amdgcn_global_load_async_to_lds_b128(
        (g4ptr)(void*)g, (l4ptr)(void*)lds, 0, 0);
#else
    *(float4*)lds = *(const float4*)g;
#endif
}
static __device__ __forceinline__ void async_copies_wait() {
#if ASYNC_LDS
    __builtin_amdgcn_s_wait_asynccnt(0);
#endif
}

// D = A(16x4) * B(4x16) + C, fp32 WMMA.
static __device__ __forceinline__ v8f wmma4(v2f a, v2f b, v8f c) {
    return __builtin_amdgcn_wmma_f32_16x16x4_f32(
        false, a, false, b, (short)0, c, false, false);
}

static __device__ __forceinline__ v8f vzero8() {
    v8f z = {0.f, 0.f, 0.f, 0.f, 0.f, 0.f, 0.f, 0.f};
    return z;
}

// ---------------------------------------------------------------------------
// Row-major fp32 GEMM:  C[MxN] = A[MxK] @ B[KxN]
// A rows [0,splitRow) from A0, rows [splitRow,M) from A1 (cat emulation).
// Block: 128 threads = 4 waves; block tile 128(M) x 64(N); K staged 16-wide.
// Wave owns 32 rows: per K-substep 2 A-frag loads + 4 B-frag loads -> 8 WMMA.
// As: row-major stride 20 (conflict-free A-frag b64 reads).
// BsT: K-major (transposed) stride 20 -> B-frag is one aligned b64 read.
// ---------------------------------------------------------------------------
__global__ __launch_bounds__(128) void gemm_f32_wmma(
    const float* __restrict__ A0, const float* __restrict__ A1, int splitRow,
    const float* __restrict__ B, float* __restrict__ C,
    int M, int N, int K)
{
    __shared__ float As[128 * 20];    // 128 M-rows x 16 K (pad 20)
    __shared__ float BsT[64 * 20];    // 64 N-cols x 16 K (transposed, pad 20)

    const int tid  = threadIdx.x;
    const int wv   = tid >> 5;
    const int lane = tid & 31;
    const int half = lane >> 4;       // 0: lanes 0-15, 1: lanes 16-31
    const int col  = lane & 15;
    const int hi   = half * 2;        // K-subcolumn select for A/B fragments
    const int n0   = blockIdx.x * 64;
    const int m0   = blockIdx.y * 128;

    v8f acc[2][4];
    #pragma unroll
    for (int rb = 0; rb < 2; ++rb)
        #pragma unroll
        for (int t = 0; t < 4; ++t) acc[rb][t] = vzero8();

    for (int k0 = 0; k0 < K; k0 += 16) {
        __syncthreads();
        // Stage A tile: 128 rows x 16 cols, 16B chunks (async when available)
        {
            int r  = tid >> 2;            // 0..31
            int c4 = (tid & 3) * 4;       // 0,4,8,12
            #pragma unroll
            for (int rr = r; rr < 128; rr += 32) {
                int grow = m0 + rr;
                const float* Ap = (grow < splitRow)
                    ? (A0 + (size_t)grow * K)
                    : (A1 + (size_t)(grow - splitRow) * K);
                copy16_to_lds(&As[rr * 20 + c4], Ap + k0 + c4);
            }
        }
        // Stage B tile transposed: 16 K-rows x 64 N-cols -> BsT[n][k]
        {
            int s = tid;
            #pragma unroll
            for (int t = 0; t < 2; ++t, s += 128) {
                int r  = s >> 4;          // 0..15 (K row)
                int c4 = (s & 15) * 4;    // 0..60 (N col group)
                float4 v = *(const float4*)(B + (size_t)(k0 + r) * N + n0 + c4);
                BsT[(c4 + 0) * 20 + r] = v.x;
                BsT[(c4 + 1) * 20 + r] = v.y;
                BsT[(c4 + 2) * 20 + r] = v.z;
                BsT[(c4 + 3) * 20 + r] = v.w;
            }
        }
        async_copies_wait();
        __syncthreads();

        #pragma unroll
        for (int kk = 0; kk < 4; ++kk) {
            // A fragments: row = col (lane&15), K = 4kk + hi + {0,1}
            v2f a0 = *(const v2f*)&As[(wv * 32 + col) * 20 + kk * 4 + hi];
            v2f a1 = *(const v2f*)&As[(wv * 32 + 16 + col) * 20 + kk * 4 + hi];
            #pragma unroll
            for (int nt = 0; nt < 4; ++nt) {
                v2f b = *(const v2f*)&BsT[(nt * 16 + col) * 20 + kk * 4 + hi];
                acc[0][nt] = wmma4(a0, b, acc[0][nt]);
                acc[1][nt] = wmma4(a1, b, acc[1][nt]);
            }
        }
    }

    // Store: C/D layout -> row = v + 8*half, col = lane&15
    #pragma unroll
    for (int rb = 0; rb < 2; ++rb) {
        #pragma unroll
        for (int nt = 0; nt < 4; ++nt) {
            #pragma unroll
            for (int v = 0; v < 8; ++v) {
                int row = rb * 16 + v + 8 * half;
                C[(size_t)(m0 + wv * 32 + row) * N + n0 + nt * 16 + col] =
                    acc[rb][nt][v];
            }
        }
    }
}

// ---------------------------------------------------------------------------
// Fused flash attention with Transformer-XL relative shift.
// Grid: (QLEN/64, BSZ, NH). Block: 128 threads = 4 waves; wave w owns
// i-rows [wgI0+16w, wgI0+16w+16). Streams j in 16-wide tiles, causally
// truncated at j <= i_max + MLEN.
//
// Rel-shift: BD[i,j] = BD_raw[i, j + QLEN-1 - i]. The block stages an 80-row
// RK window covering all 4 waves; each wave WMMAs two adjacent 16-wide
// BD_raw tiles and realigns along the diagonal with ds_bpermute (__shfl).
// ---------------------------------------------------------------------------
__global__ __launch_bounds__(128) void attn_flash_relshift(
    const float* __restrict__ QKV,   // 16384 x 1536: [Q | K | V] head-major cols
    const float* __restrict__ RK,    // 2048 x 512
    const float* __restrict__ rwb,   // 8 x 64
    const float* __restrict__ rrb,   // 8 x 64
    float* __restrict__ AV,          // 8192 x 512 (i*BSZ+b rows)
    float* __restrict__ ENT)         // 16*8*8 partials
{
    __shared__ float Kt[16 * 68];    // K tile, j-major (AC B-frags contiguous)
    __shared__ float VtT[64 * 20];   // V tile, d-major (P@V B-frags contiguous)
    __shared__ float Rt[80 * 68];    // RK window, c-major
    __shared__ float Ps[4][16 * 20]; // per-wave P staging (C-layout -> A-layout)
    __shared__ float entRed[4];

    const int tid  = threadIdx.x;
    const int wv   = tid >> 5;
    const int lane = tid & 31;
    const int half = lane >> 4;
    const int col  = lane & 15;
    const int hi   = half * 2;
    const int ib   = blockIdx.x;
    const int bb   = blockIdx.y;
    const int n    = blockIdx.z;
    const int wgI0 = ib * 64;
    const int i0   = wgI0 + wv * 16;

    // Q fragments in A layout (row = col, d = 4kk + hi + {0,1}), with biases.
    v2f q1[16], q2[16];
    {
        const float* qrow = QKV + (size_t)(MLEN + i0 + col) * (3 * DMODEL) + n * DH;
        #pragma unroll
        for (int kk = 0; kk < 16; ++kk) {
            int d  = kk * 4 + hi;
            v2f q  = *(const v2f*)(qrow + d);
            v2f bw = *(const v2f*)(rwb + n * DH + d);
            v2f br = *(const v2f*)(rrb + n * DH + d);
            q1[kk] = q + bw;
            q2[kk] = q + br;
        }
    }

    float m[8], l[8], S1[8];
    #pragma unroll
    for (int v = 0; v < 8; ++v) { m[v] = NEG_BIG; l[v] = 0.f; S1[v] = 0.f; }
    v8f acc[4];
    #pragma unroll
    for (int t = 0; t < 4; ++t) acc[t] = vzero8();

    const int o0 = 48 - wv * 16;      // wave's c-window offset inside Rt
    int jCount = wgI0 + 64 + MLEN;    // max unmasked j (exclusive), mult of 16
    if (jCount > KLEN_) jCount = KLEN_;
    const int jTiles = jCount / 16;

    for (int jt = 0; jt < jTiles; ++jt) {
        const int j0    = jt * 16;
        const int cwin0 = j0 - wgI0 + 960;   // >= 0 always
        __syncthreads();
        // Cooperative staging: K tile + RK window (async 16B chunks),
        // V tile transposed to d-major (sync scatter stores).
        {
            int r  = tid >> 3;            // 0..15 (j row)
            int c4 = (tid & 7) * 4;       // 0..28
            const float* kp = QKV + (size_t)(j0 + r) * (3 * DMODEL) + DMODEL + n * DH;
            copy16_to_lds(&Kt[r * 68 + c4],      kp + c4);
            copy16_to_lds(&Kt[r * 68 + c4 + 32], kp + c4 + 32);

            const float* vp = kp + DMODEL;
            float4 v0 = *(const float4*)(vp + c4);
            float4 v1 = *(const float4*)(vp + c4 + 32);
            VtT[(c4 + 0) * 20 + r] = v0.x;
            VtT[(c4 + 1) * 20 + r] = v0.y;
            VtT[(c4 + 2) * 20 + r] = v0.z;
            VtT[(c4 + 3) * 20 + r] = v0.w;
            VtT[(c4 + 32) * 20 + r] = v1.x;
            VtT[(c4 + 33) * 20 + r] = v1.y;
            VtT[(c4 + 34) * 20 + r] = v1.z;
            VtT[(c4 + 35) * 20 + r] = v1.w;

            #pragma unroll
            for (int rr = r; rr < 80; rr += 16) {
                int c = cwin0 + rr;
                c = (c < KLEN_ - 1) ? c : (KLEN_ - 1);   // clamped rows are masked
                const float* rp = RK + (size_t)c * DMODEL + n * DH;
                copy16_to_lds(&Rt[rr * 68 + c4],      rp + c4);
                copy16_to_lds(&Rt[rr * 68 + c4 + 32], rp + c4 + 32);
            }
        }
        async_copies_wait();
        __syncthreads();

        // AC tile + two BD_raw tiles, K-dim = 64 (16 x WMMA f32 16x16x4).
        v8f ac = vzero8(), bd0 = vzero8(), bd1 = vzero8();
        #pragma unroll
        for (int kk = 0; kk < 16; ++kk) {
            v2f bk = *(const v2f*)&Kt[col * 68 + kk * 4 + hi];
            ac  = wmma4(q1[kk], bk, ac);
            v2f b0 = *(const v2f*)&Rt[(o0 + col) * 68 + kk * 4 + hi];
            bd0 = wmma4(q2[kk], b0, bd0);
            v2f b1 = *(const v2f*)&Rt[(o0 + 16 + col) * 68 + kk * 4 + hi];
            bd1 = wmma4(q2[kk], b1, bd1);
        }

        // Diagonal realign (rel-shift), mask, online softmax + entropy stats.
        float rsv[8];
        #pragma unroll
        for (int v = 0; v < 8; ++v) {
            const int row = v + 8 * half;
            const int idx = col - row + 15;                 // 0..30
            float b0v = __shfl(bd0[v], (idx & 15) + (half << 4), 32);
            float b1v = __shfl(bd1[v], ((idx - 16) & 15) + (half << 4), 32);
            float bdv = (idx < 16) ? b0v : b1v;
            float s = (ac[v] + bdv) * SCALE;
            if (j0 + col > i0 + row + MLEN) s = NEG_BIG;    // causal(+mem) mask

            float tm = s;
            tm = fmaxf(tm, __shfl_xor(tm, 1, 32));
            tm = fmaxf(tm, __shfl_xor(tm, 2, 32));
            tm = fmaxf(tm, __shfl_xor(tm, 4, 32));
            tm = fmaxf(tm, __shfl_xor(tm, 8, 32));
            float mnew = fmaxf(m[v], tm);
            float rs   = __expf(m[v] - mnew);
            float p    = __expf(s - mnew);                  // masked -> exactly 0
            float psum = p, ps1 = p * s;                    // 0 * NEG_BIG == -0
            psum += __shfl_xor(psum, 1, 32);  ps1 += __shfl_xor(ps1, 1, 32);
            psum += __shfl_xor(psum, 2, 32);  ps1 += __shfl_xor(ps1, 2, 32);
            psum += __shfl_xor(psum, 4, 32);  ps1 += __shfl_xor(ps1, 4, 32);
            psum += __shfl_xor(psum, 8, 32);  ps1 += __shfl_xor(ps1, 8, 32);
            m[v]  = mnew;
            l[v]  = l[v]  * rs + psum;
            S1[v] = S1[v] * rs + ps1;
            rsv[v] = rs;
            Ps[wv][row * 20 + col] = p;                     // stage for P@V
        }
        #pragma unroll
        for (int dt = 0; dt < 4; ++dt)
            #pragma unroll
            for (int v = 0; v < 8; ++v) acc[dt][v] *= rsv[v];

        // acc += P(16x16) @ V(16x64): 4 d-tiles x 4 K-substeps of WMMA.
        #pragma unroll
        for (int kk = 0; kk < 4; ++kk) {
            v2f a = *(const v2f*)&Ps[wv][col * 20 + kk * 4 + hi];
            #pragma unroll
            for (int dt = 0; dt < 4; ++dt) {
                v2f bv = *(const v2f*)&VtT[(dt * 16 + col) * 20 + kk * 4 + hi];
                acc[dt] = wmma4(a, bv, acc[dt]);
            }
        }
    }

    // Normalize and store attn_vec (row = i*BSZ+b, col = n*64 + d).
    float invl[8];
    #pragma unroll
    for (int v = 0; v < 8; ++v) invl[v] = 1.0f / l[v];
    #pragma unroll
    for (int dt = 0; dt < 4; ++dt) {
        #pragma unroll
        for (int v = 0; v < 8; ++v) {
            int row = v + 8 * half;
            AV[(size_t)((i0 + row) * BSZ + bb) * DMODEL + n * DH + dt * 16 + col] =
                acc[dt][v] * invl[v];
        }
    }

    // Streaming entropy: H_row = m + ln(l) - S1/l; deterministic reduction.
    float h = 0.f;
    #pragma unroll
    for (int v = 0; v < 8; ++v) h += m[v] + logf(l[v]) - S1[v] * invl[v];
    h = (col == 0) ? h : 0.f;     // rows duplicated across 16 lanes of a half
    h += __shfl_xor(h, 1, 32);
    h += __shfl_xor(h, 2, 32);
    h += __shfl_xor(h, 4, 32);
    h += __shfl_xor(h, 8, 32);
    h += __shfl_xor(h, 16, 32);
    if (lane == 0) entRed[wv] = h;
    __syncthreads();
    if (tid == 0)
        ENT[(ib * BSZ + bb) * NH + n] = entRed[0] + entRed[1] + entRed[2] + entRed[3];
}

// Fixed-order entropy finalize: out[n] = mean over (iblock, b) partials.
__global__ __launch_bounds__(32) void ent_finalize(
    const float* __restrict__ ENT, float* __restrict__ out)
{
    int n = threadIdx.x;
    if (n < NH) {
        float s = 0.f;
        for (int ib = 0; ib < QLEN / 64; ++ib)
            for (int b = 0; b < BSZ; ++b)
                s += ENT[(ib * BSZ + b) * NH + n];
        out[n] = s * (1.0f / (float)(QLEN * BSZ));
    }
}

// ---------------------------------------------------------------------------
// Host-side launch. Workspace layout (floats):
//   QKV : 16384*1536 = 25,165,824   (96 MB)
//   RK  :  2048* 512 =  1,048,576   ( 4 MB)
//   AV  :  8192* 512 =  4,194,304   (16 MB)
//   ENT :  1024                      (4 KB)
// d_out: attn_out (1024*8*512 floats) ++ attn_entropy (8 floats).
// attn_mask input (d_in[5]) is recomputed analytically (j > i + MLEN).
// ---------------------------------------------------------------------------
extern "C" void kernel_launch(void* const* d_in, const int* in_sizes, int n_in,
                              void* d_out, int out_size, void* d_ws, size_t ws_size,
                              hipStream_t stream) {
    (void)in_sizes; (void)n_in; (void)out_size; (void)ws_size;
    const float* w    = (const float*)d_in[0];
    const float* r    = (const float*)d_in[1];
    const float* rwb  = (const float*)d_in[2];
    const float* rrb  = (const float*)d_in[3];
    const float* mems = (const float*)d_in[4];
    const float* Wqkv = (const float*)d_in[6];
    const float* Wr   = (const float*)d_in[7];
    const float* Wo   = (const float*)d_in[8];
    float* out = (float*)d_out;

    float* QKV = (float*)d_ws;
    float* RK  = QKV + (size_t)16384 * 1536;
    float* AV  = RK  + (size_t)2048 * 512;
    float* ENT = AV  + (size_t)8192 * 512;

    // K1: [mems; w] @ W_qkv  (cat row t*8+b: t<1024 -> mems, else w)
    gemm_f32_wmma<<<dim3(1536 / 64, 16384 / 128), 128, 0, stream>>>(
        mems, w, MLEN * BSZ, Wqkv, QKV, 16384, 1536, DMODEL);
    // K2: r @ W_r
    gemm_f32_wmma<<<dim3(512 / 64, 2048 / 128), 128, 0, stream>>>(
        r, r, KLEN_, Wr, RK, KLEN_, DMODEL, DMODEL);
    // K3: fused attention
    attn_flash_relshift<<<dim3(QLEN / 64, BSZ, NH), 128, 0, stream>>>(
        QKV, RK, rwb, rrb, AV, ENT);
    // K4: attn_vec @ W_o -> d_out
    gemm_f32_wmma<<<dim3(512 / 64, 8192 / 128), 128, 0, stream>>>(
        AV, AV, QLEN * BSZ, Wo, out, QLEN * BSZ, DMODEL, DMODEL);
    // K5: entropy
    ent_finalize<<<1, 32, 0, stream>>>(ENT, out + (size_t)QLEN * BSZ * DMODEL);
}